// ImageCube_43387759624716
// MI455X (gfx1250) — compile-verified
//
#include <hip/hip_runtime.h>

#define NCHAN 4
#define NPIX  2048
#define NXB   1025                  // NPIX/2 + 1
#define NVIS  100000
#define KNZ   36
#define NGRID (NPIX * NXB)
#define SCALE 2.5e-5f               // (CELL_RAD^2 / ARCSEC^2) = 0.005^2

typedef float v2f __attribute__((ext_vector_type(2)));
typedef float v8f __attribute__((ext_vector_type(8)));
typedef unsigned int u32x4 __attribute__((ext_vector_type(4)));
typedef unsigned int u32x8 __attribute__((ext_vector_type(8)));

#define PI_F 3.14159265358979323846f

// ---------------------------------------------------------------------------
// Tensor Data Mover: 2D tile load Global -> LDS (D# groups 0/1 per CDNA5 ISA
// ch.8; VADDR2/VADDR3 omitted -> NULL -> 2D descriptor). Issued by one wave,
// tracked with TENSORcnt. Inline asm keeps it portable across toolchains.
// ---------------------------------------------------------------------------
__device__ __forceinline__ unsigned int lds_offset_of(const void* p)
{
    // generic pointers to LDS: addr[31:0] is the LDS byte offset
    return (unsigned int)(uintptr_t)p;
}

__device__ __forceinline__ void tdm_load_2d(unsigned int lds_byte_off, const void* gptr,
                                            unsigned int tile_d0, unsigned int tile_d1,
                                            unsigned int tensor_d0, unsigned int tensor_d1,
                                            unsigned int stride0_elems)
{
    const unsigned long long ga = (unsigned long long)(uintptr_t)gptr;
    u32x4 g0;
    g0[0] = 0x1u;                                              // count=1, user D#
    g0[1] = lds_byte_off;                                      // lds_addr
    g0[2] = (unsigned int)ga;                                  // global_addr[31:0]
    g0[3] = (unsigned int)((ga >> 32) & 0x01FFFFFFu)           // global_addr[56:32]
          | (2u << 30);                                        // type=2 ("image")
    u32x8 g1;
    g1[0] = 2u << 16;                                          // wg_mask=0, data_size=4B
    g1[1] = (tensor_d0 & 0xFFFFu) << 16;                       // tensor_dim0[15:0]
    g1[2] = ((tensor_d0 >> 16) & 0xFFFFu)                      // tensor_dim0[31:16]
          | ((tensor_d1 & 0xFFFFu) << 16);                     // tensor_dim1[15:0]
    g1[3] = ((tensor_d1 >> 16) & 0xFFFFu)                      // tensor_dim1[31:16]
          | ((tile_d0 & 0xFFFFu) << 16);                       // tile_dim0
    g1[4] = tile_d1 & 0xFFFFu;                                 // tile_dim1 (tile_dim2=0)
    g1[5] = stride0_elems;                                     // tensor_dim0_stride[31:0]
    g1[6] = 0u;                                                // stride0[47:32], stride1[15:0]
    g1[7] = 0u;                                                // stride1[47:16]
    asm volatile("tensor_load_to_lds %0, %1" :: "s"(g0), "s"(g1) : "memory");
}

// ---------------------------------------------------------------------------
// Radix-16 DIF stage computed with V_WMMA_F32_16X16X4_F32.
//   Y(16x16) = W16(16x16) * X(16x16), complex:
//     Yr = Wr*Xr + (-Wi)*Xi ; Yi = Wr*Xi + Wi*Xr      (16 WMMAs / stage / wave)
// One wave owns 16 butterflies (the 16 columns of the tile); 8 waves cover
// all 128 radix-16 butterflies of a 2048-point line. In-place in LDS; a
// wave reads/writes only its own butterflies, so barriers only between stages.
// ---------------------------------------------------------------------------
template <int L, int M>
__device__ __forceinline__ void radix16_stage_wmma(float* sre, float* sim, int tid)
{
    const int lane = tid & 31;
    const int wave = tid >> 5;
    const int lo   = lane & 15;
    const int hi   = (lane >> 4) & 1;

    const int bf    = wave * 16 + lo;
    const int blk   = bf / M;
    const int j     = bf - blk * M;
    const int ebase = blk * L + j;

    v8f yr = {0.f, 0.f, 0.f, 0.f, 0.f, 0.f, 0.f, 0.f};
    v8f yi = {0.f, 0.f, 0.f, 0.f, 0.f, 0.f, 0.f, 0.f};

#pragma unroll
    for (int kb = 0; kb < 4; ++kb) {
        // A fragments: W[m][r] = exp(-2*pi*i*m*r/16), m = lo
        const int rx = kb * 4 + (hi ? 2 : 0);
        const int ry = rx + 1;
        float s0, c0, s1, c1;
        __sincosf(-(PI_F / 8.0f) * (float)(lo * rx), &s0, &c0);
        __sincosf(-(PI_F / 8.0f) * (float)(lo * ry), &s1, &c1);
        v2f aR, aI, aIn;
        aR.x = c0;  aR.y = c1;
        aI.x = s0;  aI.y = s1;
        aIn.x = -s0; aIn.y = -s1;       // F32 WMMA has no A-negate modifier

        // B fragments: X[r][bf] gathered from LDS
        const int bkx = kb * 4 + hi;
        const int bky = kb * 4 + 2 + hi;
        v2f bR, bI;
        bR.x = sre[ebase + M * bkx];  bI.x = sim[ebase + M * bkx];
        bR.y = sre[ebase + M * bky];  bI.y = sim[ebase + M * bky];

        yr = __builtin_amdgcn_wmma_f32_16x16x4_f32(false, aR,  false, bR, (short)0, yr, false, false);
        yr = __builtin_amdgcn_wmma_f32_16x16x4_f32(false, aIn, false, bI, (short)0, yr, false, false);
        yi = __builtin_amdgcn_wmma_f32_16x16x4_f32(false, aR,  false, bI, (short)0, yi, false, false);
        yi = __builtin_amdgcn_wmma_f32_16x16x4_f32(false, aI,  false, bR, (short)0, yi, false, false);
    }

    // twiddle W_L^(j*k) and in-place store
#pragma unroll
    for (int i = 0; i < 8; ++i) {
        const int k = i + hi * 8;
        float ts, tc;
        __sincosf(-(2.0f * PI_F) * (float)(j * k) / (float)L, &ts, &tc);
        const float r  = yr[i];
        const float im = yi[i];
        sre[ebase + M * k] = r * tc - im * ts;
        sim[ebase + M * k] = r * ts + im * tc;
    }
}

// Final radix-8 DIF stage: 256 length-8 DFTs, one per thread (VALU).
__device__ __forceinline__ void radix8_stage(float* sre, float* sim, int tid)
{
    const float W8R[8] = {1.f, 0.70710678f, 0.f, -0.70710678f, -1.f, -0.70710678f, 0.f, 0.70710678f};
    const float W8I[8] = {0.f, -0.70710678f, -1.f, -0.70710678f, 0.f, 0.70710678f, 1.f, 0.70710678f};
    const int base = tid * 8;
    float xr[8], xi[8];
#pragma unroll
    for (int r = 0; r < 8; ++r) { xr[r] = sre[base + r]; xi[r] = sim[base + r]; }
#pragma unroll
    for (int k = 0; k < 8; ++k) {
        float ar = 0.f, ai = 0.f;
#pragma unroll
        for (int r = 0; r < 8; ++r) {
            const float wr = W8R[(k * r) & 7];
            const float wi = W8I[(k * r) & 7];
            ar += xr[r] * wr - xi[r] * wi;
            ai += xr[r] * wi + xi[r] * wr;
        }
        sre[base + k] = ar;
        sim[base + k] = ai;
    }
}

// 2048-point complex FFT in LDS, 256 threads. Result digit-reversed:
// LDS index a*128 + b*8 + c holds bin c*256 + b*16 + a  (a,b in [0,16), c in [0,8)).
__device__ __forceinline__ void fft2048(float* sre, float* sim, int tid)
{
    radix16_stage_wmma<2048, 128>(sre, sim, tid);
    __syncthreads();
    radix16_stage_wmma<128, 8>(sre, sim, tid);
    __syncthreads();
    radix8_stage(sre, sim, tid);
    __syncthreads();
}

__device__ __forceinline__ int digitrev_idx(int f)
{
    const int a = f & 15, b = (f >> 4) & 15, c = f >> 8;   // f = c*256 + b*16 + a
    return a * 128 + b * 8 + c;
}

// ---------------------------------------------------------------------------
// Kernel A: taper + row rFFT. One block per (c, y) row; keep bins 0..1024.
// Row staged into LDS by the Tensor Data Mover (contiguous 2048x1 tile).
// ---------------------------------------------------------------------------
__global__ void __launch_bounds__(256)
rowfft_kernel(const float* __restrict__ cube, const float* __restrict__ corrfun,
              float* __restrict__ gr, float* __restrict__ gi)
{
    __shared__ float sre[NPIX];
    __shared__ float sim[NPIX];
    const int row = blockIdx.x;                 // 0 .. NCHAN*NPIX-1
    const int c   = row / NPIX;
    const int y   = row - c * NPIX;
    const float* src = cube + (size_t)row * NPIX;
    const float* cf  = corrfun + (size_t)y * NPIX;

    // async DMA of the raw row into sim[], then taper pass builds sre/zeros sim
    if (threadIdx.x == 0)
        tdm_load_2d(lds_offset_of(sim), src, NPIX, 1, NPIX, 1, NPIX);
    __builtin_amdgcn_s_wait_tensorcnt(0);
    __syncthreads();

    for (int n = threadIdx.x; n < NPIX; n += 256) {
        const float v = sim[n];
        sre[n] = v * cf[n] * SCALE;
        sim[n] = 0.f;
    }
    __syncthreads();

    fft2048(sre, sim, threadIdx.x);

    float* orr = gr + (size_t)c * NGRID + (size_t)y * NXB;
    float* ori = gi + (size_t)c * NGRID + (size_t)y * NXB;
    for (int f = threadIdx.x; f < NXB; f += 256) {
        const int idx = digitrev_idx(f);
        orr[f] = sre[idx];
        ori[f] = sim[idx];
    }
}

// ---------------------------------------------------------------------------
// Kernel B: in-place column FFT of gr/gi. One block per (c, x) column.
// The strided column (stride NXB floats) is gathered into contiguous LDS by
// the TDM (1 x 2048 tile, tensor_dim0_stride = NXB) — ideal DMA use; grid is
// L2-resident (67 MB < 192 MB).
// ---------------------------------------------------------------------------
__global__ void __launch_bounds__(256)
colfft_kernel(float* __restrict__ gr, float* __restrict__ gi)
{
    __shared__ float sre[NPIX];
    __shared__ float sim[NPIX];
    const int col = blockIdx.x;                 // 0 .. NCHAN*NXB-1
    const int c   = col / NXB;
    const int x   = col - c * NXB;
    float* pr = gr + (size_t)c * NGRID + x;
    float* pi = gi + (size_t)c * NGRID + x;

    if (threadIdx.x == 0) {
        tdm_load_2d(lds_offset_of(sre), pr, 1, NPIX, 1, NPIX, NXB);
        tdm_load_2d(lds_offset_of(sim), pi, 1, NPIX, 1, NPIX, NXB);
    }
    __builtin_amdgcn_s_wait_tensorcnt(0);
    __syncthreads();

    fft2048(sre, sim, threadIdx.x);

    for (int f = threadIdx.x; f < NPIX; f += 256) {
        const int idx = digitrev_idx(f);
        pr[(size_t)f * NXB] = sre[idx];
        pi[(size_t)f * NXB] = sim[idx];
    }
}

// ---------------------------------------------------------------------------
// Kernel C: sparse gather-dot. One wave32 per (c, v); lanes stream k
// coalesced, gathers hit L2, wave shuffle reduction.
// ---------------------------------------------------------------------------
__global__ void __launch_bounds__(256)
gather_dot_kernel(const float* __restrict__ w_re, const float* __restrict__ w_im,
                  const int* __restrict__ cols_re, const int* __restrict__ cols_im,
                  const float* __restrict__ gr, const float* __restrict__ gi,
                  float* __restrict__ out_re, float* __restrict__ out_im)
{
    const int gw = blockIdx.x * (blockDim.x >> 5) + (threadIdx.x >> 5);  // (c,v) flat
    if (gw >= NCHAN * NVIS) return;
    const int lane = threadIdx.x & 31;
    const int c    = gw / NVIS;
    const size_t base = (size_t)gw * KNZ;
    const float* vr = gr + (size_t)c * NGRID;
    const float* vi = gi + (size_t)c * NGRID;

    // warm L2/L0 for the next visibility's weight stream (global_prefetch_b8)
    __builtin_prefetch(w_re + base + KNZ, 0, 1);
    __builtin_prefetch(w_im + base + KNZ, 0, 1);

    float accr = 0.f, acci = 0.f;
    for (int k = lane; k < KNZ; k += 32) {
        accr += w_re[base + k] * vr[cols_re[base + k]];
        acci += w_im[base + k] * vi[cols_im[base + k]];
    }
#pragma unroll
    for (int off = 16; off > 0; off >>= 1) {
        accr += __shfl_xor(accr, off, 32);
        acci += __shfl_xor(acci, off, 32);
    }
    if (lane == 0) {
        out_re[gw] = accr;
        out_im[gw] = acci;
    }
}

// ---------------------------------------------------------------------------
extern "C" void kernel_launch(void* const* d_in, const int* in_sizes, int n_in,
                              void* d_out, int out_size, void* d_ws, size_t ws_size,
                              hipStream_t stream)
{
    (void)in_sizes; (void)n_in; (void)out_size; (void)ws_size;
    const float* cube    = (const float*)d_in[0];
    const float* corrfun = (const float*)d_in[1];
    const float* w_re    = (const float*)d_in[2];
    const float* w_im    = (const float*)d_in[3];
    const int*   cols_re = (const int*)d_in[4];
    const int*   cols_im = (const int*)d_in[5];

    float* gr = (float*)d_ws;                            // [NCHAN][NGRID] real grid
    float* gi = gr + (size_t)NCHAN * NGRID;              // [NCHAN][NGRID] imag grid

    float* out_re = (float*)d_out;                       // [NCHAN][NVIS]
    float* out_im = out_re + (size_t)NCHAN * NVIS;       // [NCHAN][NVIS]

    rowfft_kernel<<<NCHAN * NPIX, 256, 0, stream>>>(cube, corrfun, gr, gi);
    colfft_kernel<<<NCHAN * NXB, 256, 0, stream>>>(gr, gi);

    const int waves  = NCHAN * NVIS;
    const int blocks = (waves + 7) / 8;                  // 8 waves per 256-thread block
    gather_dot_kernel<<<blocks, 256, 0, stream>>>(w_re, w_im, cols_re, cols_im,
                                                  gr, gi, out_re, out_im);
}